// SpatialGraphSubsystem_3865470566685
// MI455X (gfx1250) — compile-verified
//
#include <hip/hip_runtime.h>
#include <hip/hip_bf16.h>

typedef __attribute__((ext_vector_type(2))) float v2f;
typedef __attribute__((ext_vector_type(4))) float v4f;
typedef __attribute__((ext_vector_type(8))) float v8f;

#define NUM_E 36
#define DIM   256              // D == N == 256
#define BPB   4                // batches per block
#define ROWS  (BPB * 12)       // 48 node rows per block
#define LDST  260              // LDS row stride (dwords): 256 + 4 pad; row base = 1040B (16B aligned)

__device__ __forceinline__ float softplus_f(float x) {
    // numerically stable log1p(exp(x))
    return fmaxf(x, 0.0f) + log1pf(__expf(-fabsf(x)));
}

__global__ __launch_bounds__(256)
void fused_graph_node_gemm(const float* __restrict__ edge_feats,  // [B,36,256]
                           const float* __restrict__ edge_w,      // [36]
                           const float* __restrict__ W,           // [256,256] row-major [out,in]
                           const float* __restrict__ bias,        // [256]
                           float* __restrict__ out,               // [B,12,256]
                           int B)
{
    __shared__ float G[ROWS * LDST];   // 48 x 260 fp32 ~= 49.9 KB
    __shared__ float wsm[NUM_E];

    const int tid = threadIdx.x;
    const int b0  = blockIdx.x * BPB;
    const bool full_block = (b0 + BPB <= B);   // uniform: no per-lane guards in fast path

    if (tid < NUM_E) wsm[tid] = softplus_f(edge_w[tid]);
    __syncthreads();

    // ---- Stage 1: build G = concat(node_u, node_v). Each thread: 1 batch (tid>>6),
    //      4 consecutive columns ((tid&63)*4) -> b128 global loads, b128 LDS stores. ----
    {
        const int bb = tid >> 6;          // batch within block, 0..3
        const int c4 = (tid & 63) * 4;    // first column of this thread's group
        if (b0 + bb < B) {
            v4f au[6] = {};
            v4f av[6] = {};
            const float* ef = edge_feats + ((size_t)(b0 + bb) * NUM_E) * DIM + c4;
#pragma unroll
            for (int e = 0; e < NUM_E; ++e) {
                v4f v = *(const v4f*)(ef + (size_t)e * DIM) * wsm[e];
                au[e / 6] += v;   // node_u[i] <- sum over j
                av[e % 6] += v;   // node_v[j] <- sum over i
            }
#pragma unroll
            for (int n = 0; n < 6; ++n) {
                *(v4f*)&G[(bb * 12 + n)     * LDST + c4] = au[n];
                *(v4f*)&G[(bb * 12 + 6 + n) * LDST + c4] = av[n];
            }
        }
    }
    __syncthreads();

    // ---- Stage 2: out[48x256] = relu(G x W^T + b) via V_WMMA_F32_16X16X4_F32 ----
    const int wave = tid >> 5;
    const int lane = tid & 31;
    const int l15  = lane & 15;
    const int hi   = lane >> 4;       // 0: K pair {0,1}, 1: K pair {2,3}

    for (int t = 0; t < 6; ++t) {
        const int tile = wave * 6 + t;    // 0..47
        const int rt   = tile >> 4;       // 0..2  (16-row tile)
        const int ct   = tile & 15;       // 0..15 (16-col tile)
        const int m0   = rt * 16;
        const int n0   = ct * 16;

        const float* Arow = &G[(m0 + l15) * LDST];           // A: G row for this lane
        const float* Brow = W + (size_t)(n0 + l15) * DIM;    // B[k][n] = W[n][k]

        v8f acc = {};
#pragma unroll 4
        for (int k = 0; k < DIM; k += 4) {
            v2f a  = *(const v2f*)(Arow + k + 2 * hi);
            v2f bm = *(const v2f*)(Brow + k + 2 * hi);
            acc = __builtin_amdgcn_wmma_f32_16x16x4_f32(false, a, false, bm,
                                                        (short)0, acc, false, false);
        }

        // D layout: vgpr r -> global out row (b0*12 + m0 + hi*8 + r), col n0 + l15.
        // Rows are contiguous in the output, so the 8 stores are base + r*DIM.
        const float bn   = bias[n0 + l15];
        const int   lr0  = m0 + hi * 8;                      // first local row for this lane
        float* orow = out + ((size_t)b0 * 12 + lr0) * DIM + n0 + l15;

        if (full_block) {
#pragma unroll
            for (int r = 0; r < 8; ++r)
                orow[(size_t)r * DIM] = fmaxf(acc[r] + bn, 0.0f);
        } else {
            const int maxr = (B - b0) * 12 - lr0;            // rows remaining for this lane
#pragma unroll
            for (int r = 0; r < 8; ++r)
                if (r < maxr)
                    orow[(size_t)r * DIM] = fmaxf(acc[r] + bn, 0.0f);
        }
    }
}

__global__ void write_softplus_weights(const float* __restrict__ edge_w,
                                       float* __restrict__ out_w)
{
    int i = threadIdx.x;
    if (i < NUM_E) out_w[i] = softplus_f(edge_w[i]);
}

extern "C" void kernel_launch(void* const* d_in, const int* in_sizes, int n_in,
                              void* d_out, int out_size, void* d_ws, size_t ws_size,
                              hipStream_t stream) {
    const float* edge_feats = (const float*)d_in[0];
    const float* edge_w     = (const float*)d_in[1];
    const float* W          = (const float*)d_in[2];
    const float* bias       = (const float*)d_in[3];
    float* out = (float*)d_out;

    const int B = in_sizes[0] / (NUM_E * DIM);

    const int blocks = (B + BPB - 1) / BPB;
    fused_graph_node_gemm<<<blocks, 256, 0, stream>>>(edge_feats, edge_w, W, bias, out, B);

    // weights appended after node_feats in the flat output
    float* out_w = out + (size_t)B * 12 * DIM;
    write_softplus_weights<<<1, 64, 0, stream>>>(edge_w, out_w);
}